// KrylovNPLR_57604101374535
// MI455X (gfx1250) — compile-verified
//
#include <hip/hip_runtime.h>
#include <math.h>

// ---------------------------------------------------------------------------
// S4 / NPLR convolution kernel K_L on MI455X (gfx1250).
// Stage 1: Cauchy resolvent + Woodbury per (h, l)    -> A_re, A_imneg in ws
// Stage 2: inverse real DFT as f32 WMMA GEMM:
//          out = A_re @ COS + A_imneg @ SIN          (M=256, K=4112, N=8192)
// ---------------------------------------------------------------------------

static constexpr int kH      = 256;
static constexpr int kNH     = 32;          // NHALF
static constexpr int kL      = 8192;
static constexpr int kLh     = 4097;        // L/2 + 1
static constexpr int kKPAD   = 4112;        // kLh rounded up to multiple of 16
static constexpr int kKCHUNK = 16;          // K columns staged in LDS per sync
static constexpr float kTwoPiOverL = 7.6699039394282067e-4f;  // 2*pi/8192

typedef float v2f __attribute__((ext_vector_type(2)));
typedef float v8f __attribute__((ext_vector_type(8)));

// ---------------------------------------------------------------------------
// Stage 1: one thread per (h, l).  Computes
//   kf = (r00 - r01*r10/(1+r11)) * 2/(1+node),  r_ij = dt * sum_n [ v_ij/(z-w)
//        + conj(v_ij)/(z-conj(w)) ],  z = 2(1-node)/(1+node),  node = e^{-2πil/L}
// then stores irfft-folded weights:
//   A_re = c_l * Re(kf),  A_imneg = -c_l * Im(kf),  c_l = {1,2}/L.
// ---------------------------------------------------------------------------
__global__ __launch_bounds__(256) void s4_cauchy_kernel(
    const float* __restrict__ log_dt, const float* __restrict__ w_re,
    const float* __restrict__ w_im, const float* __restrict__ BV_re,
    const float* __restrict__ BV_im, const float* __restrict__ CV_re,
    const float* __restrict__ CV_im, float* __restrict__ Are,
    float* __restrict__ Asn)
{
  int idx = blockIdx.x * 256 + threadIdx.x;
  if (idx >= kH * kKPAD) return;
  int h = idx / kKPAD;
  int l = idx - h * kKPAD;
  if (l >= kLh) { Are[idx] = 0.0f; Asn[idx] = 0.0f; return; }

  float dt = __expf(log_dt[h]);

  // node = exp(-2*pi*i*l/L)
  float phi = -kTwoPiOverL * (float)l;
  float nr, ni;
  __sincosf(phi, &ni, &nr);

  // z = 2*(1-node)/(1+node)  (purely imaginary in exact math)
  float dr = 1.0f + nr, di = ni;
  float inv_den = __frcp_rn(dr * dr + di * di);
  float zr = 2.0f * ((1.0f - nr) * dr - ni * di) * inv_den;
  float zi = 2.0f * (-ni * dr - (1.0f - nr) * di) * inv_den;

  float r00r = 0.f, r00i = 0.f, r01r = 0.f, r01i = 0.f;
  float r10r = 0.f, r10i = 0.f, r11r = 0.f, r11i = 0.f;

  const float* wrp = w_re + h * kNH;
  const float* wip = w_im + h * kNH;
  const float* bvr = BV_re + h * 2 * kNH;   // rows [B; p]
  const float* bvi = BV_im + h * 2 * kNH;
  const float* cvr = CV_re + h * 2 * kNH;   // rows [C; q]
  const float* cvi = CV_im + h * 2 * kNH;

#pragma unroll 4
  for (int n = 0; n < kNH; ++n) {
    float wr = wrp[n] * dt, wi = wip[n] * dt;
    // p = 1/(z - w)
    float ar = zr - wr, ai = zi - wi;
    float s1 = __frcp_rn(ar * ar + ai * ai);
    float pr = ar * s1, pi = -ai * s1;
    // q = 1/(z - conj(w))
    float a2r = zr - wr, a2i = zi + wi;
    float s2 = __frcp_rn(a2r * a2r + a2i * a2i);
    float qr = a2r * s2, qi = -a2i * s2;

    // v*p + conj(v)*q:  re += vr*(pr+qr) + vi*(qi-pi)
    //                   im += vr*(pi+qi) + vi*(pr-qr)
    float Sr = pr + qr, Dq = qi - pi, Si = pi + qi, Dp = pr - qr;

    float b0r = bvr[n],        b0i = bvi[n];
    float b1r = bvr[kNH + n],  b1i = bvi[kNH + n];
    float c0r = cvr[n],        c0i = cvi[n];
    float c1r = cvr[kNH + n],  c1i = cvi[kNH + n];

    { float vr = c0r * b0r - c0i * b0i, vi = c0r * b0i + c0i * b0r;
      r00r += vr * Sr + vi * Dq; r00i += vr * Si + vi * Dp; }
    { float vr = c0r * b1r - c0i * b1i, vi = c0r * b1i + c0i * b1r;
      r01r += vr * Sr + vi * Dq; r01i += vr * Si + vi * Dp; }
    { float vr = c1r * b0r - c1i * b0i, vi = c1r * b0i + c1i * b0r;
      r10r += vr * Sr + vi * Dq; r10i += vr * Si + vi * Dp; }
    { float vr = c1r * b1r - c1i * b1i, vi = c1r * b1i + c1i * b1r;
      r11r += vr * Sr + vi * Dq; r11i += vr * Si + vi * Dp; }
  }

  r00r *= dt; r00i *= dt; r01r *= dt; r01i *= dt;
  r10r *= dt; r10i *= dt; r11r *= dt; r11i *= dt;

  // Woodbury: kf = r00 - r01*r10/(1+r11)
  float tr = r01r * r10r - r01i * r10i;
  float ti = r01r * r10i + r01i * r10r;
  float er = 1.0f + r11r, ei = r11i;
  float se = __frcp_rn(er * er + ei * ei);
  float kr = r00r - (tr * er + ti * ei) * se;
  float ki = r00i - (ti * er - tr * ei) * se;

  // kf *= 2/(1+node)
  float fr = 2.0f * dr * inv_den, fi = -2.0f * di * inv_den;
  float kfr = kr * fr - ki * fi;
  float kfi = kr * fi + ki * fr;

  float cl = (l == 0 || l == kL / 2) ? (1.0f / kL) : (2.0f / kL);
  Are[idx] = cl * kfr;
  Asn[idx] = -cl * kfi;
}

// ---------------------------------------------------------------------------
// Stage 2: out = A_re @ COS + A_imneg @ SIN via V_WMMA_F32_16X16X4_F32.
// One block = 512 threads = 16 waves = all 256 H rows of one 64-wide N strip.
// A 16x64 cos/sin twiddle panel (K-chunk = 16) is generated cooperatively
// into LDS once per sync window and shared by all 16 waves.
// Exact angles: m = (l*t) & 8191, theta = m * 2pi/8192.
// cos-GEMM and sin-GEMM use SEPARATE accumulators (summed in the epilogue)
// so no two dependent WMMAs are adjacent -> no accumulator copy ping-pong,
// hazard slots filled with independent WMMAs.
// ---------------------------------------------------------------------------
__global__ __launch_bounds__(512) void s4_idft_wmma_kernel(
    const float* __restrict__ Are, const float* __restrict__ Asn,
    float* __restrict__ out)
{
  __shared__ __attribute__((aligned(16))) float ldsC[64][kKCHUNK];
  __shared__ __attribute__((aligned(16))) float ldsS[64][kKCHUNK];

  const int tid   = threadIdx.x;
  const int lane  = tid & 31;
  const int wave  = tid >> 5;
  const int nbase = blockIdx.x * 64;
  const int hbase = wave * 16;

  // A-tile addressing (ISA 16x4 f32 A layout): lanes 0-15 rows, K pair by half
  const int arow = hbase + (lane & 15);
  const int koff = (lane >> 4) * 2;
  const float* aRePtr = Are + (size_t)arow * kKPAD + koff;
  const float* aImPtr = Asn + (size_t)arow * kKPAD + koff;

  v8f accC[4] = {};  // cos-path accumulators: 4 N-subtiles of 16x16 f32
  v8f accS[4] = {};  // sin-path accumulators

  for (int kc = 0; kc < kKPAD; kc += kKCHUNK) {
    __syncthreads();
    // 16(K) x 64(N) twiddle panel: 1024 angles, 2 per thread
#pragma unroll
    for (int j = 0; j < 2; ++j) {
      int idx = tid + 512 * j;
      int gk = idx & (kKCHUNK - 1);
      int gn = idx >> 4;
      int m = ((kc + gk) * (nbase + gn)) & (kL - 1);
      float th = (float)m * kTwoPiOverL;
      float sv, cv;
      __sincosf(th, &sv, &cv);
      ldsC[gn][gk] = cv;
      ldsS[gn][gk] = sv;
    }
    __syncthreads();

#pragma unroll
    for (int kk = 0; kk < kKCHUNK; kk += 4) {
      v2f aRe = *(const v2f*)(aRePtr + kc + kk);
      v2f aIm = *(const v2f*)(aImPtr + kc + kk);
#pragma unroll
      for (int s = 0; s < 4; ++s) {
        int nl = s * 16 + (lane & 15);
        v2f bc = *(const v2f*)&ldsC[nl][kk + koff];  // B 4x16: K pair per half
        v2f bs = *(const v2f*)&ldsS[nl][kk + koff];
        accC[s] = __builtin_amdgcn_wmma_f32_16x16x4_f32(
            false, aRe, false, bc, (short)0, accC[s], false, false);
        accS[s] = __builtin_amdgcn_wmma_f32_16x16x4_f32(
            false, aIm, false, bs, (short)0, accS[s], false, false);
      }
    }
  }

  // D layout: VGPR r -> M = r (lanes 0-15) or r+8 (lanes 16-31), N = lane&15
#pragma unroll
  for (int s = 0; s < 4; ++s) {
#pragma unroll
    for (int r = 0; r < 8; ++r) {
      int row = hbase + r + (lane >> 4) * 8;
      int col = nbase + s * 16 + (lane & 15);
      out[(size_t)row * kL + col] = accC[s][r] + accS[s][r];
    }
  }
}

// ---------------------------------------------------------------------------
extern "C" void kernel_launch(void* const* d_in, const int* in_sizes, int n_in,
                              void* d_out, int out_size, void* d_ws,
                              size_t ws_size, hipStream_t stream) {
  (void)in_sizes; (void)n_in; (void)out_size; (void)ws_size;
  const float* log_dt = (const float*)d_in[0];
  const float* w_re   = (const float*)d_in[1];
  const float* w_im   = (const float*)d_in[2];
  const float* BV_re  = (const float*)d_in[3];
  const float* BV_im  = (const float*)d_in[4];
  const float* CV_re  = (const float*)d_in[5];
  const float* CV_im  = (const float*)d_in[6];
  // d_in[7] is L (scalar); kL hardcoded to match the reference.

  float* Are = (float*)d_ws;                 // 256 x 4112 f32
  float* Asn = Are + (size_t)kH * kKPAD;     // 256 x 4112 f32  (~8.4 MB total)
  float* out = (float*)d_out;

  int total = kH * kKPAD;
  s4_cauchy_kernel<<<(total + 255) / 256, 256, 0, stream>>>(
      log_dt, w_re, w_im, BV_re, BV_im, CV_re, CV_im, Are, Asn);
  s4_idft_wmma_kernel<<<kL / 64, 512, 0, stream>>>(Are, Asn, out);
}